// StructuralImportanceAttention_15040975470958
// MI455X (gfx1250) — compile-verified
//
#include <hip/hip_runtime.h>
#include <hip/hip_bf16.h>
#include <math.h>

typedef __attribute__((ext_vector_type(2))) float v2f;
typedef __attribute__((ext_vector_type(8))) float v8f;

#define HID 256   // hidden dim
#define PD  64    // projection dim

// ---------------------------------------------------------------------------
// Kernel 0: zero the H-dependent scratch region.
// hreg layout (floats): [ hefeat 64H | segK 64H | counts H | smax H | denom H ]
// ---------------------------------------------------------------------------
__global__ void k_zero(float* __restrict__ hreg, const int* __restrict__ pH) {
    long long H = (long long)(*pH);
    long long total = H * 131LL;
    long long stride = (long long)gridDim.x * blockDim.x;
    for (long long i = (long long)blockIdx.x * blockDim.x + threadIdx.x;
         i < total; i += stride)
        hreg[i] = 0.0f;
}

// ---------------------------------------------------------------------------
// Kernel 1: per-NODE projection  K = feats @ Wk^T,  V = feats @ Wv^T
// W staged in LDS in a pair-major swizzled layout:
//   lds[(k>>1)*128 + ((col + 16*(k>>1)) & 63)*2 + (k&1)] = W[col][k]
// -> conflict-free ds_load_b64 per B fragment; lane-halves (kp, kp+1)
//    land on complementary bank sets.
// Each wave owns TWO 16-row blocks: 8 independent WMMAs per k-step, 2 per B load.
// Store epilogue: wave-uniform fast path (unconditional stores) when the
// whole 32-row block is in range; masked path only for the tail block.
// ---------------------------------------------------------------------------
__device__ __forceinline__ void proj_phase(const float* __restrict__ feats,
                                           const float* __restrict__ blds,
                                           float* __restrict__ dst,
                                           int N, int rowbase, int half, int r)
{
    const v8f z = {0.f,0.f,0.f,0.f,0.f,0.f,0.f,0.f};
    v8f acc[2][4];
#pragma unroll
    for (int s = 0; s < 2; ++s)
#pragma unroll
        for (int t = 0; t < 4; ++t) acc[s][t] = z;

    int row0 = rowbase + r;       if (row0 >= N) row0 = N - 1;
    int row1 = rowbase + 16 + r;  if (row1 >= N) row1 = N - 1;
    const size_t a0 = (size_t)row0 * HID;
    const size_t a1 = (size_t)row1 * HID;

#pragma unroll 2
    for (int k0 = 0; k0 < HID; k0 += 4) {
        const int kp = (k0 >> 1) + half;            // K-pair index for this lane
        v2f fa0 = *(const v2f*)(feats + a0 + k0 + 2 * half);
        v2f fa1 = *(const v2f*)(feats + a1 + k0 + 2 * half);
#pragma unroll
        for (int t = 0; t < 4; ++t) {
            const int col = t * 16 + r;
            const int sc  = (col + (kp << 4)) & 63; // bank swizzle
            v2f b = *(const v2f*)(blds + kp * 128 + sc * 2);
            acc[0][t] = __builtin_amdgcn_wmma_f32_16x16x4_f32(
                false, fa0, false, b, (short)0, acc[0][t], false, false);
            acc[1][t] = __builtin_amdgcn_wmma_f32_16x16x4_f32(
                false, fa1, false, b, (short)0, acc[1][t], false, false);
        }
    }

    // D layout: element i of lane -> row = i + 8*half, col = lane%16
    if (rowbase + 32 <= N) {
        // fast path: one per-lane base, immediate offsets, no masking
        float* base = dst + (size_t)(rowbase + 8 * half) * PD + r;
#pragma unroll
        for (int s = 0; s < 2; ++s)
#pragma unroll
            for (int t = 0; t < 4; ++t)
#pragma unroll
                for (int i = 0; i < 8; ++i)
                    base[(size_t)(s * 16 + i) * PD + t * 16] = acc[s][t][i];
    } else {
#pragma unroll
        for (int s = 0; s < 2; ++s)
#pragma unroll
            for (int t = 0; t < 4; ++t)
#pragma unroll
                for (int i = 0; i < 8; ++i) {
                    const int orow = rowbase + s * 16 + i + 8 * half;
                    if (orow < N)
                        dst[(size_t)orow * PD + t * 16 + r] = acc[s][t][i];
                }
    }
}

__global__ void __launch_bounds__(256)
k_kv_gemm(const float* __restrict__ feats, const float* __restrict__ Wk,
          const float* __restrict__ Wv, float* __restrict__ Knode,
          float* __restrict__ Vnode, int N)
{
    __shared__ float blds[PD * HID];        // 16384 floats = 64 KB
    const int tid  = threadIdx.x;
    const int lane = tid & 31;
    const int wave = tid >> 5;
    const int half = lane >> 4;
    const int r    = lane & 15;
    const int rowbase = (blockIdx.x * 8 + wave) * 32;

    // ---- phase K ----
    for (int i = tid; i < PD * HID; i += 256) {       // Wk row-major [col][k]
        const int col = i >> 8, k = i & 255;
        const int kp = k >> 1;
        blds[kp * 128 + (((col + (kp << 4)) & 63) << 1) + (k & 1)] = Wk[i];
    }
    __syncthreads();
    proj_phase(feats, blds, Knode, N, rowbase, half, r);
    __syncthreads();

    // ---- phase V ----
    for (int i = tid; i < PD * HID; i += 256) {
        const int col = i >> 8, k = i & 255;
        const int kp = k >> 1;
        blds[kp * 128 + (((col + (kp << 4)) & 63) << 1) + (k & 1)] = Wv[i];
    }
    __syncthreads();
    proj_phase(feats, blds, Vnode, N, rowbase, half, r);
}

// ---------------------------------------------------------------------------
// Kernel 2: scatter-add keys into per-hyperedge sums + membership counts
// ---------------------------------------------------------------------------
__global__ void k_scatter_k(const float* __restrict__ Knode,
                            const int* __restrict__ nidx,
                            const int* __restrict__ heidx,
                            float* __restrict__ hreg,
                            const int* __restrict__ pH, int M)
{
    const long long H = (long long)(*pH);
    float* segK   = hreg + 64LL * H;
    float* counts = hreg + 128LL * H;
    const long long total  = (long long)M * PD;
    const long long stride = (long long)gridDim.x * blockDim.x;
    for (long long i = (long long)blockIdx.x * blockDim.x + threadIdx.x;
         i < total; i += stride) {
        const int m = (int)(i >> 6);
        const int p = (int)(i & 63);
        const int n  = nidx[m];
        const int he = heidx[m];
        atomicAdd(&segK[(size_t)he * PD + p], Knode[(size_t)n * PD + p]);
        if (p == 0) atomicAdd(&counts[he], 1.0f);
    }
}

// ---------------------------------------------------------------------------
// Kernel 3: L2 distance to own centroid, score = d/sqrt(P); segment max
// (scores >= 0 -> bitwise uint atomicMax on zero-initialized slots is exact;
//  empty segments keep 0, matching the reference's isfinite fix)
// ---------------------------------------------------------------------------
__global__ void k_dist(const float* __restrict__ Knode,
                       const int* __restrict__ nidx,
                       const int* __restrict__ heidx,
                       float* __restrict__ hreg,
                       float* __restrict__ scores,
                       const int* __restrict__ pH, int M)
{
    const long long H = (long long)(*pH);
    const float* segK   = hreg + 64LL * H;
    const float* counts = hreg + 128LL * H;
    unsigned*    smax   = (unsigned*)(hreg + 129LL * H);
    const long long stride = (long long)gridDim.x * blockDim.x;
    for (long long m = (long long)blockIdx.x * blockDim.x + threadIdx.x;
         m < M; m += stride) {
        const int n  = nidx[m];
        const int he = heidx[m];
        const float invc = 1.0f / fmaxf(counts[he], 1.0f);
        const float4* kv = (const float4*)(Knode + (size_t)n  * PD);
        const float4* cv = (const float4*)(segK  + (size_t)he * PD);
        float acc = 0.0f;
#pragma unroll 4
        for (int q = 0; q < PD / 4; ++q) {
            const float4 kk = kv[q];
            const float4 cc = cv[q];
            const float dx = kk.x - cc.x * invc;
            const float dy = kk.y - cc.y * invc;
            const float dz = kk.z - cc.z * invc;
            const float dw = kk.w - cc.w * invc;
            acc += dx * dx + dy * dy + dz * dz + dw * dw;
        }
        const float score = sqrtf(acc) * 0.125f;   // 1/sqrt(64)
        scores[m] = score;
        atomicMax(&smax[he], __float_as_uint(score));
    }
}

// ---------------------------------------------------------------------------
// Kernel 4: exp(score - segmax), scatter-add the softmax denominator
// ---------------------------------------------------------------------------
__global__ void k_exp(const int* __restrict__ heidx,
                      float* __restrict__ hreg,
                      float* __restrict__ scores,
                      const int* __restrict__ pH, int M)
{
    const long long H = (long long)(*pH);
    const unsigned* smax = (const unsigned*)(hreg + 129LL * H);
    float* denom = hreg + 130LL * H;
    const long long stride = (long long)gridDim.x * blockDim.x;
    for (long long m = (long long)blockIdx.x * blockDim.x + threadIdx.x;
         m < M; m += stride) {
        const int he = heidx[m];
        const float e = __expf(scores[m] - __uint_as_float(smax[he]));
        scores[m] = e;                 // scores now holds exp'd values
        atomicAdd(&denom[he], e);
    }
}

// ---------------------------------------------------------------------------
// Kernel 5: he_feats += attn * V   (attn = exp / denom)
// ---------------------------------------------------------------------------
__global__ void k_weighted(const float* __restrict__ Vnode,
                           const int* __restrict__ nidx,
                           const int* __restrict__ heidx,
                           float* __restrict__ hreg,
                           const float* __restrict__ scores,
                           const int* __restrict__ pH, int M)
{
    const long long H = (long long)(*pH);
    float* hefeat = hreg;                       // region base
    const float* denom = hreg + 130LL * H;
    const long long total  = (long long)M * PD;
    const long long stride = (long long)gridDim.x * blockDim.x;
    for (long long i = (long long)blockIdx.x * blockDim.x + threadIdx.x;
         i < total; i += stride) {
        const int m = (int)(i >> 6);
        const int p = (int)(i & 63);
        const int n  = nidx[m];
        const int he = heidx[m];
        const float attn = scores[m] / denom[he];
        atomicAdd(&hefeat[(size_t)he * PD + p],
                  attn * Vnode[(size_t)n * PD + p]);
    }
}

// ---------------------------------------------------------------------------
// Kernel 6: out[H,256] = he_feats[H,64] @ Wv[64,256]
// Wv staged in LDS pair-major + swizzle:
//   lds[(k>>1)*512 + ((col + 16*(k>>1)) & 255)*2 + (k&1)] = Wv[k][col]
// Persistent WMMA waves grid-stride over (rowblk, colgroup) work items
// since H is only known on the device. Fast-path epilogue when the whole
// 16-row block is in range.
// ---------------------------------------------------------------------------
__global__ void __launch_bounds__(256)
k_out_gemm(const float* __restrict__ hreg, const float* __restrict__ Wv,
           float* __restrict__ out, const int* __restrict__ pH)
{
    __shared__ float blds[PD * HID];        // 16384 floats = 64 KB
    const long long H = (long long)(*pH);
    const float* hefeat = hreg;
    const int tid  = threadIdx.x;
    const int lane = tid & 31;
    const int wave = tid >> 5;
    const int half = lane >> 4;
    const int r    = lane & 15;

    for (int i = tid; i < PD * HID; i += 256) {       // Wv row-major [k][col]
        const int k = i >> 8, col = i & 255;
        const int kp = k >> 1;
        blds[kp * 512 + (((col + (kp << 4)) & 255) << 1) + (k & 1)] = Wv[i];
    }
    __syncthreads();

    const long long nRowBlk = (H + 15) >> 4;
    const long long nWork   = nRowBlk * 4;      // 4 col-groups of 64 cols
    const long long wstride = (long long)gridDim.x * 8;

    for (long long w = (long long)blockIdx.x * 8 + wave; w < nWork; w += wstride) {
        const long long rowblk = w >> 2;
        const int       cg     = (int)(w & 3);

        long long row = rowblk * 16 + r;
        if (row >= H) row = H - 1;              // clamp loads, mask stores

        const v8f z = {0.f,0.f,0.f,0.f,0.f,0.f,0.f,0.f};
        v8f acc[4];
#pragma unroll
        for (int t = 0; t < 4; ++t) acc[t] = z;

        const size_t arow = (size_t)row * PD;
#pragma unroll 2
        for (int k0 = 0; k0 < PD; k0 += 4) {
            const int kp = (k0 >> 1) + half;
            v2f a = *(const v2f*)(hefeat + arow + k0 + 2 * half);
#pragma unroll
            for (int t = 0; t < 4; ++t) {
                const int col = cg * 64 + t * 16 + r;
                const int sc  = (col + (kp << 4)) & 255;
                v2f b = *(const v2f*)(blds + kp * 512 + sc * 2);
                acc[t] = __builtin_amdgcn_wmma_f32_16x16x4_f32(
                    false, a, false, b, (short)0, acc[t], false, false);
            }
        }

        if ((rowblk + 1) * 16 <= H) {
            // fast path: unconditional stores, immediate offsets
            float* base = out + (size_t)(rowblk * 16 + 8 * half) * HID
                              + cg * 64 + r;
#pragma unroll
            for (int t = 0; t < 4; ++t)
#pragma unroll
                for (int i = 0; i < 8; ++i)
                    base[(size_t)i * HID + t * 16] = acc[t][i];
        } else {
#pragma unroll
            for (int t = 0; t < 4; ++t)
#pragma unroll
                for (int i = 0; i < 8; ++i) {
                    const long long orow = rowblk * 16 + i + 8 * half;
                    if (orow < H)
                        out[(size_t)orow * HID + cg * 64 + t * 16 + r] =
                            acc[t][i];
                }
        }
    }
}

// ---------------------------------------------------------------------------
extern "C" void kernel_launch(void* const* d_in, const int* in_sizes, int n_in,
                              void* d_out, int out_size, void* d_ws, size_t ws_size,
                              hipStream_t stream) {
    const float* feats = (const float*)d_in[0];   // [N,256]
    const int*   nidx  = (const int*)d_in[1];     // [M]
    const int*   heidx = (const int*)d_in[2];     // [M]
    const int*   pH    = (const int*)d_in[3];     // scalar on device
    const float* Wk    = (const float*)d_in[4];   // [64,256]
    const float* Wv    = (const float*)d_in[5];   // [64,256]
    float* out = (float*)d_out;

    const int N = in_sizes[0] / HID;
    const int M = in_sizes[1];

    // Workspace layout (floats):
    //   [ Knode 64N | Vnode 64N | scores M | hreg 131H ]
    float* ws     = (float*)d_ws;
    float* Knode  = ws;
    float* Vnode  = ws + (size_t)64 * N;
    float* scores = ws + (size_t)128 * N;
    float* hreg   = ws + (size_t)128 * N + (size_t)M;

    // 0) zero H-dependent scratch
    k_zero<<<2048, 256, 0, stream>>>(hreg, pH);

    // 1) per-node K/V projection (WMMA f32, LDS-staged weights)
    k_kv_gemm<<<(N + 255) / 256, 256, 0, stream>>>(feats, Wk, Wv,
                                                   Knode, Vnode, N);

    // 2) centroid scatter-sum + counts
    k_scatter_k<<<4096, 256, 0, stream>>>(Knode, nidx, heidx, hreg, pH, M);

    // 3) distances / scores + segment max
    k_dist<<<4096, 256, 0, stream>>>(Knode, nidx, heidx, hreg, scores, pH, M);

    // 4) exp + denominator
    k_exp<<<4096, 256, 0, stream>>>(heidx, hreg, scores, pH, M);

    // 5) attention-weighted value scatter
    k_weighted<<<4096, 256, 0, stream>>>(Vnode, nidx, heidx, hreg, scores,
                                         pH, M);

    // 6) back-projection to hidden dim (WMMA f32, LDS-staged weights)
    k_out_gemm<<<640, 256, 0, stream>>>(hreg, Wv, out, pH);
}